// Shift_Block_split3_Up_45758581572311
// MI455X (gfx1250) — compile-verified
//
#include <hip/hip_runtime.h>
#include <stdint.h>

// ---------------- problem geometry ----------------
#define IMG_H 512
#define IMG_W 512
#define DIL   2
#define TILE_H 32
#define TILE_W 64
#define SMH (TILE_H + 2 * DIL)   // 36 rows incl. halo
#define SMP 72                   // LDS window width == row stride (floats), 16B-multiple
#define CPR (SMP / 4)            // 18 16-byte chunks per row
#define NCHK (SMH * CPR)         // 648 chunks staged per tile
#define NTH 256                  // 8 wave32s per block

typedef __attribute__((address_space(1))) int glb_i32_t;  // global (AS1)
typedef __attribute__((address_space(3))) int lds_i32_t;  // LDS (AS3)
typedef int v4i __attribute__((vector_size(16)));         // 16B chunk type
typedef __attribute__((address_space(1))) v4i glb_v4i_t;
typedef __attribute__((address_space(3))) v4i lds_v4i_t;
typedef float v2f __attribute__((ext_vector_type(2)));
typedef float v4f __attribute__((ext_vector_type(4)));

// ---- CDNA5 async global->LDS copies (ASYNCcnt-tracked) ----
__device__ __forceinline__ void async_copy_b128(const float* gp, float* lp) {
#if __has_builtin(__builtin_amdgcn_global_load_async_to_lds_b128)
  __builtin_amdgcn_global_load_async_to_lds_b128(
      (glb_v4i_t*)(uintptr_t)gp, (lds_v4i_t*)(uint32_t)(uintptr_t)lp,
      /*offset=*/0, /*cpol=*/0);
#elif __has_builtin(__builtin_amdgcn_global_load_async_to_lds_b32)
#pragma unroll
  for (int i = 0; i < 4; ++i)
    __builtin_amdgcn_global_load_async_to_lds_b32(
        (glb_i32_t*)(uintptr_t)(gp + i),
        (lds_i32_t*)(uint32_t)(uintptr_t)(lp + i), 0, 0);
#else
  unsigned lofs = (unsigned)(uintptr_t)lp;
  asm volatile("global_load_async_to_lds_b128 %0, %1, off"
               :
               : "v"(lofs), "v"(gp)
               : "memory");
#endif
}

__device__ __forceinline__ void wait_async_zero() {
#if __has_builtin(__builtin_amdgcn_s_wait_asynccnt)
  __builtin_amdgcn_s_wait_asynccnt(0);
#else
  asm volatile("s_wait_asynccnt 0" ::: "memory");
#endif
}

__device__ __forceinline__ float fast_sigmoid(float t) {
#if __has_builtin(__builtin_amdgcn_exp2f)
  return 1.0f / (1.0f + __builtin_amdgcn_exp2f(t * -1.4426950408889634f));
#else
  return 1.0f / (1.0f + exp2f(t * -1.4426950408889634f));
#endif
}

__global__ __launch_bounds__(NTH) void shift_block_up_kernel(
    const float* __restrict__ x, const float* __restrict__ y,
    const float* __restrict__ w1, const float* __restrict__ b1,
    const float* __restrict__ w2, const float* __restrict__ b2,
    float* __restrict__ out, int C) {
  __shared__ float sm[SMH * SMP];

  const int tid = threadIdx.x;
  const int plane = blockIdx.y;        // b*C + c
  const int c = plane % C;
  const int tile = blockIdx.x;         // 8 tiles across, 16 down
  const int tx0 = (tile & 7) * TILE_W;
  const int ty0 = (tile >> 3) * TILE_H;

  const float* xp = x + (size_t)plane * (IMG_H * IMG_W);
  const v2f* yp = (const v2f*)(y + (size_t)plane * (4 * IMG_H * IMG_W));
  v2f* op = (v2f*)(out + (size_t)plane * (4 * IMG_H * IMG_W));

  // ---- Phase 1: async-stage raw x window (+halo) into LDS, 16B chunks ----
  // Window cols [tx0-4, tx0+68): 16B-aligned in global and LDS.
  // Chunks clamped into the plane only ever cover all-OOB slots (zeroed in
  // phase 2), so clamping is fault-safe and never corrupts live data.
  const int PLANE_MAX = IMG_H * IMG_W - 4;
  for (int e = tid; e < NCHK; e += NTH) {
    const int r = e / CPR;
    const int cc = (e - r * CPR) * 4;       // col within window
    const int gy = ty0 - DIL + r;
    const int gcol = tx0 - 4 + cc;
    int off = gy * IMG_W + gcol;
    off = min(max(off, 0), PLANE_MAX);
    async_copy_b128(xp + off, &sm[r * SMP + cc]);
  }
  wait_async_zero();
  __syncthreads();

  const float w1c = w1[c], b1c = b1[c];
  const float w2c = w2[c], b2c = b2[c];

  // ---- Phase 2: in-place activation (v4f RMW); zero OOB slots ----
  for (int e = tid; e < NCHK; e += NTH) {
    const int r = e / CPR;
    const int cc = (e - r * CPR) * 4;
    const int gy = ty0 - DIL + r;
    const int gx0c = tx0 - 4 + cc;
    const bool rowok = (unsigned)gy < IMG_H;
    v4f v = *(const v4f*)&sm[r * SMP + cc];
    v4f o;
    o.x = (rowok && (unsigned)(gx0c + 0) < IMG_W)
              ? fmaxf(fmaf(v.x, w1c, b1c), 0.0f) : 0.0f;
    o.y = (rowok && (unsigned)(gx0c + 1) < IMG_W)
              ? fmaxf(fmaf(v.y, w1c, b1c), 0.0f) : 0.0f;
    o.z = (rowok && (unsigned)(gx0c + 2) < IMG_W)
              ? fmaxf(fmaf(v.z, w1c, b1c), 0.0f) : 0.0f;
    o.w = (rowok && (unsigned)(gx0c + 3) < IMG_W)
              ? fmaxf(fmaf(v.w, w1c, b1c), 0.0f) : 0.0f;
    *(v4f*)&sm[r * SMP + cc] = o;
  }
  __syncthreads();

  // ---- Phase 3: 9-tap dilated box-sum, gate, 2x nearest upsample-add ----
  const int tx = tid & (TILE_W - 1);  // 0..63
  const int tyq = tid >> 6;           // 0..3
#pragma unroll
  for (int k = 0; k < TILE_H / 4; ++k) {
    const int ty = tyq * (TILE_H / 4) + k;
    // source pixel (ty0+ty, tx0+tx) sits at LDS [ty+2][tx+4]
    const float* r0 = &sm[(ty + 0) * SMP + tx + 2];
    const float* r2 = &sm[(ty + 2) * SMP + tx + 2];
    const float* r4 = &sm[(ty + 4) * SMP + tx + 2];
    const float xo = r2[2];
    float s = r0[0] + r0[2] + r0[4];
    s += r2[0] + xo + r2[4];
    s += r4[0] + r4[2] + r4[4];
    const float o = fast_sigmoid(fmaf(s, w2c, b2c)) * xo;

    const int gy2 = (ty0 + ty) * 2;       // upsampled row
    const int gxc = tx0 + tx;             // float2 column in 1024-wide row
    const size_t i0 = (size_t)gy2 * IMG_W + gxc;        // IMG_W float2 per row
    const size_t i1 = i0 + IMG_W;
    v2f a0 = __builtin_nontemporal_load(yp + i0);
    v2f a1 = __builtin_nontemporal_load(yp + i1);
    a0.x += o; a0.y += o;
    a1.x += o; a1.y += o;
    __builtin_nontemporal_store(a0, op + i0);
    __builtin_nontemporal_store(a1, op + i1);
  }
}

extern "C" void kernel_launch(void* const* d_in, const int* in_sizes, int n_in,
                              void* d_out, int out_size, void* d_ws,
                              size_t ws_size, hipStream_t stream) {
  (void)n_in; (void)out_size; (void)d_ws; (void)ws_size;
  const float* x = (const float*)d_in[0];
  const float* y = (const float*)d_in[1];
  const float* w1 = (const float*)d_in[2];
  const float* b1 = (const float*)d_in[3];
  const float* w2 = (const float*)d_in[4];
  const float* b2 = (const float*)d_in[5];
  float* out = (float*)d_out;

  const int C = in_sizes[2];                              // 3 channels
  const int planes = in_sizes[0] / (IMG_H * IMG_W);       // B*C = 24
  const int tiles = (IMG_W / TILE_W) * (IMG_H / TILE_H);  // 8*16 = 128

  dim3 grid(tiles, planes);
  shift_block_up_kernel<<<grid, NTH, 0, stream>>>(x, y, w1, b1, w2, b2, out, C);
}